// SelectionHead_17420387353203
// MI455X (gfx1250) — compile-verified
//
#include <hip/hip_runtime.h>

// SelectionHead forward for MI455X (gfx1250, wave32).
// B=4 rows, N=32768, K=1000 gumbel-top-k iterations.
// One workgroup (1024 threads = 32 wave32 waves) per row:
//   scores (log2-domain, max-shifted once) -> 32 VGPRs/lane,
//   khot -> 128 KB LDS (of 320 KB/WGP).
// The 1000-step loop is a serial latency chain: exactly ONE barrier and one
// block-wide sum per step (chained v_wmma_f32_16x16x4_f32 reductions), raw
// v_exp_f32/v_log_f32 (base-2 domain, no ln2 conversions, no denorm fixups),
// and a single v_rcp_f32 for the softmax normalization.

#define N_ELE 32768
#define KSEL  1000
#define TPB   1024
#define EPT   (N_ELE / TPB)   // 32 elements per thread
#define NW    (TPB / 32)      // 32 wave32 waves
#define EPS_TINY 1.1754943508222875e-38f  // np.finfo(float32).tiny (= FLT_MIN)
#define NEG_HUGE (-3.402823466e+38f)
#define LOG2E 1.44269504088896340736f

typedef __attribute__((ext_vector_type(2))) float v2f;
typedef __attribute__((ext_vector_type(8))) float v8f;

struct BlockScratch {
  float red[2][NW];   // double-buffered block_sum scratch (parity per call)
  float mred[NW];     // block_max scratch (disjoint, cold path)
  float mbc;
  int   tie;
};

// Raw hardware transcendentals (hot loop only; args are never denormal).
__device__ __forceinline__ float fast_exp2(float x) {
#if defined(__gfx1250__) && __has_builtin(__builtin_amdgcn_exp2f)
  return __builtin_amdgcn_exp2f(x);     // v_exp_f32
#else
  return exp2f(x);
#endif
}
__device__ __forceinline__ float fast_log2(float x) {
#if defined(__gfx1250__) && __has_builtin(__builtin_amdgcn_logf)
  return __builtin_amdgcn_logf(x);      // v_log_f32 (arg >= FLT_MIN: no fixup)
#else
  return log2f(x);
#endif
}
__device__ __forceinline__ float fast_rcp(float x) {
#if defined(__gfx1250__) && __has_builtin(__builtin_amdgcn_rcpf)
  return __builtin_amdgcn_rcpf(x);      // v_rcp_f32, not the IEEE div chain
#else
  return 1.0f / x;
#endif
}

// ---- full 32-lane sum via two chained WMMAs (no LDS, no shuffles) --------
// WMMA1: A[m][0]=x(m), A[m][2]=x(m+16), B=ones -> D[m][n]=x(m)+x(m+16).
// Summing the 8 D VGPRs: lanes 0-15 all hold S_A, lanes 16-31 all hold S_B.
// WMMA2 on h: D2[m][n] = h(m)+h(m+16) = S_A+S_B = full 32-lane sum in every
// lane of d2[0].
__device__ __forceinline__ float wave_sum(float x) {
#if defined(__gfx1250__) && __has_builtin(__builtin_amdgcn_wmma_f32_16x16x4_f32)
  v2f a;    a[0] = x;       a[1] = 0.0f;
  v2f ones; ones[0] = 1.0f; ones[1] = 1.0f;   // all-ones is layout-invariant
  v8f c = {};
  v8f d = __builtin_amdgcn_wmma_f32_16x16x4_f32(
      false, a, false, ones, (short)0, c, false, false);
  float h = ((d[0] + d[1]) + (d[2] + d[3])) + ((d[4] + d[5]) + (d[6] + d[7]));
  v2f a2; a2[0] = h; a2[1] = 0.0f;
  v8f d2 = __builtin_amdgcn_wmma_f32_16x16x4_f32(
      false, a2, false, ones, (short)0, c, false, false);
  return d2[0];
#else
  #pragma unroll
  for (int off = 16; off > 0; off >>= 1) x += __shfl_xor(x, off, 32);
  return x;
#endif
}

__device__ __forceinline__ float wave_max(float x) {   // cold path only
  #pragma unroll
  for (int off = 16; off > 0; off >>= 1) x = fmaxf(x, __shfl_xor(x, off, 32));
  return x;
}

// ONE-barrier block sum: wave partials -> red[ph], barrier, then EVERY wave
// redundantly reduces all 32 partials with the 2-WMMA wave_sum (identical
// result in all lanes; no exec-mask dance, no broadcast slot, no 2nd barrier).
// Callers must alternate `ph` between consecutive calls: the single barrier of
// call n+1 (other buffer) orders call n's reads before call n+2's writes.
__device__ __forceinline__ float block_sum(float x, BlockScratch* s, int ph) {
  x = wave_sum(x);
  const int lane = threadIdx.x & 31;
  const int wv   = threadIdx.x >> 5;
  if (lane == 0) s->red[ph][wv] = x;
  __syncthreads();
  return wave_sum(s->red[ph][lane]);   // conflict-free: lane i -> bank i
}

__device__ __forceinline__ float block_max(float x, BlockScratch* s) {
  x = wave_max(x);
  const int lane = threadIdx.x & 31;
  const int wv   = threadIdx.x >> 5;
  if (lane == 0) s->mred[wv] = x;
  __syncthreads();
  if (wv == 0) {
    float y = wave_max(s->mred[lane]);
    if (lane == 0) s->mbc = y;
  }
  __syncthreads();
  return s->mbc;
}

__global__ __launch_bounds__(TPB)
void selection_head_kernel(const float* __restrict__ logits,
                           const float* __restrict__ gumbel,
                           float* __restrict__ out_values,
                           float* __restrict__ out_logprobs,
                           float* __restrict__ out_actions) {
  extern __shared__ float kh[];          // N_ELE floats = 128 KB dynamic LDS
  __shared__ BlockScratch s;

  const int b = blockIdx.x;
  const int t = threadIdx.x;
  const float* lg = logits + (size_t)b * N_ELE;
  const float* gm = gumbel + (size_t)b * N_ELE;

  __builtin_prefetch(lg + t, 0, 0);      // global_prefetch_b8
  __builtin_prefetch(gm + t, 0, 0);

  // scores live in registers (fully unrolled indexing); khot in LDS.
  float sc[EPT];
  #pragma unroll
  for (int j = 0; j < EPT; ++j) {
    const int i = t + j * TPB;
    sc[j]  = lg[i] + gm[i];
    kh[i]  = 0.0f;                       // thread-private LDS slots, no race
  }
  __syncthreads();

  // One-time max shift: scores are monotonically non-increasing over the
  // iterations, so exp(sc - m0) <= 1 for the whole run (same stability as
  // per-iteration max-subtracted softmax). Then convert to the log2 domain:
  // p = 2^(sc2)/sum(2^(sc2)) == e^(sc)/sum(e^(sc)), and the score update
  // becomes sc2 += log2(1-p) -- raw v_exp_f32/v_log_f32, no ln2 multiplies.
  float m0 = NEG_HUGE;
  #pragma unroll
  for (int j = 0; j < EPT; ++j) m0 = fmaxf(m0, sc[j]);
  m0 = block_max(m0, &s);
  #pragma unroll
  for (int j = 0; j < EPT; ++j) sc[j] = (sc[j] - m0) * LOG2E;

  // ---- K iterations of: p = softmax(scores); khot += p; scores += log(1-p)
  for (int it = 0; it < KSEL; ++it) {
    float s0 = 0.0f, s1 = 0.0f, s2 = 0.0f, s3 = 0.0f;  // break add chains
    #pragma unroll
    for (int j = 0; j < EPT; j += 4) {
      s0 += fast_exp2(sc[j + 0]);
      s1 += fast_exp2(sc[j + 1]);
      s2 += fast_exp2(sc[j + 2]);
      s3 += fast_exp2(sc[j + 3]);
    }
    const float ssum = block_sum((s0 + s1) + (s2 + s3), &s, it & 1);
    const float inv  = fast_rcp(ssum);

    #pragma unroll
    for (int j = 0; j < EPT; ++j) {
      const int i = t + j * TPB;
      const float p = fast_exp2(sc[j]) * inv;
      kh[i] += p;
      sc[j] += fast_log2(fmaxf(1.0f - p, EPS_TINY));  // arg >= FLT_MIN
    }
  }
  // K-loop ended at parity (KSEL-1)&1 == 1; next call uses parity 0.

  // khot -> registers (sc[] is dead); the 31-step threshold search then runs
  // entirely out of VGPRs.
  float khr[EPT];
  #pragma unroll
  for (int j = 0; j < EPT; ++j) khr[j] = kh[t + j * TPB];

  // ---- hard top-K of khot via monotone bit binary search on the threshold.
  // khot >= 0 so float bits are order-preserving as uints.
  unsigned ans = 0u;
  for (int bit = 30; bit >= 0; --bit) {   // parity bit&1: 0,1,0,... ends at 0
    const unsigned cand = ans | (1u << bit);
    const float thr = __uint_as_float(cand);
    float cnt = 0.0f;
    #pragma unroll
    for (int j = 0; j < EPT; ++j) cnt += (khr[j] >= thr) ? 1.0f : 0.0f;
    cnt = block_sum(cnt, &s, bit & 1);   // exact: counts <= 2^15 << 2^24
    if (cnt >= (float)KSEL) ans = cand;  // uniform across block
  }
  const float thrv = __uint_as_float(ans);

  float cg = 0.0f;
  #pragma unroll
  for (int j = 0; j < EPT; ++j) cg += (khr[j] > thrv) ? 1.0f : 0.0f;
  cg = block_sum(cg, &s, 1);             // prev call was parity 0
  const int need = KSEL - (int)cg;       // ties to accept (>= 0 by maximality)

  if (t == 0) s.tie = 0;
  __syncthreads();

  float act[EPT];
  #pragma unroll
  for (int j = 0; j < EPT; ++j) {
    float a = 0.0f;
    if (khr[j] > thrv) {
      a = 1.0f;
    } else if (khr[j] == thrv) {
      const int pos = atomicAdd(&s.tie, 1);
      if (pos < need) a = 1.0f;          // exactly K ones total
    }
    act[j] = a;
  }

  // ---- values = sigmoid(max(logits)); logprobs = log_softmax(logits)*actions
  // (cold path: exact e-base math)
  float lmax = NEG_HUGE;
  #pragma unroll
  for (int j = 0; j < EPT; ++j) lmax = fmaxf(lmax, lg[t + j * TPB]);
  lmax = block_max(lmax, &s);
  if (t == 0) out_values[b] = 1.0f / (1.0f + __expf(-lmax));

  float e0 = 0.0f, e1 = 0.0f, e2 = 0.0f, e3 = 0.0f;
  #pragma unroll
  for (int j = 0; j < EPT; j += 4) {
    e0 += __expf(lg[t + (j + 0) * TPB] - lmax);
    e1 += __expf(lg[t + (j + 1) * TPB] - lmax);
    e2 += __expf(lg[t + (j + 2) * TPB] - lmax);
    e3 += __expf(lg[t + (j + 3) * TPB] - lmax);
  }
  const float es = block_sum((e0 + e1) + (e2 + e3), &s, 0);  // prev sum was 1
  const float lz = lmax + __logf(es);

  #pragma unroll
  for (int j = 0; j < EPT; ++j) {
    const int i = t + j * TPB;
    const float lp = lg[i] - lz;
    out_logprobs[(size_t)b * N_ELE + i] = lp * act[j];
    out_actions[(size_t)b * N_ELE + i]  = act[j];
  }
}

extern "C" void kernel_launch(void* const* d_in, const int* in_sizes, int n_in,
                              void* d_out, int out_size, void* d_ws, size_t ws_size,
                              hipStream_t stream) {
  const float* logits = (const float*)d_in[0];
  const float* gumbel = (const float*)d_in[1];
  const int B = in_sizes[0] / N_ELE;     // 4
  float* out       = (float*)d_out;
  float* values    = out;                          // [B]
  float* logprobs  = out + B;                      // [B, N]
  float* actions   = out + B + (size_t)B * N_ELE;  // [B, N]
  const size_t shmem = (size_t)N_ELE * sizeof(float);  // 128 KB (<320 KB/WGP)
  selection_head_kernel<<<dim3(B), dim3(TPB), shmem, stream>>>(
      logits, gumbel, values, logprobs, actions);
}